// Model_25211458028159
// MI455X (gfx1250) — compile-verified
//
#include <hip/hip_runtime.h>
#include <math.h>

typedef __attribute__((ext_vector_type(2))) float f32x2;
typedef __attribute__((ext_vector_type(8))) float f32x8;
typedef __attribute__((ext_vector_type(4))) unsigned int u32x4;
typedef __attribute__((ext_vector_type(8))) int i32x8;
typedef __attribute__((ext_vector_type(4))) int i32x4;

// D = A(16x4) * B(4x16) + C, fp32 on the CDNA5 WMMA pipe.
#define WMMA_F32(a, b, c) \
  __builtin_amdgcn_wmma_f32_16x16x4_f32(false, (a), false, (b), (short)0, (c), false, false)

// Tensor Data Mover availability (device pass only; host pass falls back).
#if defined(__has_builtin)
#if __has_builtin(__builtin_amdgcn_tensor_load_to_lds) && \
    __has_builtin(__builtin_amdgcn_s_wait_tensorcnt)
#define USE_TDM 1
#endif
#endif
#ifndef USE_TDM
#define USE_TDM 0
#endif

#define XCT_STRIDE 2052   // 2052 % 64 == 4 -> conflict-free A-operand rows
#define BC_STRIDE  136    // 128 + 8 dw pad -> lanes 0-15 / 16-31 hit disjoint banks

#if USE_TDM
// DMA one 64-row x 128-col f32 chunk of W_eff into LDS, padding 8 DWORDs
// every 128 DWORDs (row stride becomes BC_STRIDE=136 in LDS).
__device__ __forceinline__ void tdm_load_chunk(const float* gsrc, unsigned int lds_byte_off)
{
  unsigned long long ga = (unsigned long long)(uintptr_t)gsrc;
  u32x4 g0;
  g0[0] = 1u;                                                 // count=1 (user D#)
  g0[1] = lds_byte_off;                                       // lds_addr
  g0[2] = (unsigned int)ga;                                   // global_addr[31:0]
  g0[3] = (unsigned int)((ga >> 32) & 0x1FFFFFFu) | (2u << 30); // addr[56:32] | type=2

  i32x8 g1;
  g1[0] = (int)((2u << 16)      // data_size = 4 bytes
              | (1u << 20)      // pad_enable
              | (6u << 22)      // pad_interval: 128 DWORDs
              | (7u << 25));    // pad_amount:   8 DWORDs
  g1[1] = (int)(128u << 16);    // abar=0 | tensor_dim0[15:0] = 128
  g1[2] = (int)(2048u << 16);   // tensor_dim0[31:16]=0 | tensor_dim1[15:0] = 2048
  g1[3] = (int)(128u << 16);    // tensor_dim1[31:16]=0 | tile_dim0 = 128
  g1[4] = (int)64u;             // tile_dim1 = 64 | tile_dim2 = 0
  g1[5] = (int)128u;            // tensor_dim0_stride = 128
  g1[6] = 0;
  g1[7] = 0;

  i32x4 z4 = {0, 0, 0, 0};
#if __clang_major__ >= 23
  i32x8 z8 = {0, 0, 0, 0, 0, 0, 0, 0};
  __builtin_amdgcn_tensor_load_to_lds(g0, g1, z4, z4, z8, 0);
#else
  __builtin_amdgcn_tensor_load_to_lds(g0, g1, z4, z4, 0);
#endif
}
#endif

// ---- shared conv(k=5,s=2,SAME: pad_lo=1) + relu + LN(32) row ------------
__device__ __forceinline__ void conv_relu_ln_row(
    const float* __restrict__ X, const float* Kc, const float* cb,
    const float* s1, const float* b1, int b, int t, int c, float* out)
{
  float x[5];
#pragma unroll
  for (int kt = 0; kt < 5; ++kt) {
    int ti = 2 * t - 1 + kt;
    x[kt] = (ti >= 0 && ti < 4096) ? X[(b * 4096 + ti) * 64 + c] : 0.0f;
  }
  float sum = 0.f, sq = 0.f;
#pragma unroll
  for (int f = 0; f < 32; ++f) {
    float v = cb[f];
#pragma unroll
    for (int kt = 0; kt < 5; ++kt) v = fmaf(x[kt], Kc[kt * 32 + f], v);
    v = v > 0.f ? v : 0.f;
    out[f] = v;
    sum += v;
    sq = fmaf(v, v, sq);
  }
  float mu  = sum * (1.f / 32.f);
  float var = sq * (1.f / 32.f) - mu * mu;
  float rs  = rsqrtf(var + 1e-6f);
#pragma unroll
  for (int f = 0; f < 32; ++f) out[f] = (out[f] - mu) * rs * s1[f] + b1[f];
}

// ---- K1: deterministic partial channel sums -----------------------------
__global__ void k1_chan(const float* __restrict__ X, const float* __restrict__ conv_k,
                        const float* __restrict__ conv_b, const float* __restrict__ ln1_s,
                        const float* __restrict__ ln1_b, float* __restrict__ chanPart)
{
  __shared__ float Kc[160], cb[32], s1[32], b1[32];
  __shared__ float part[4][64];
  int tid = threadIdx.x;
  for (int i = tid; i < 160; i += 256) Kc[i] = conv_k[i];
  if (tid < 32) { cb[tid] = conv_b[tid]; s1[tid] = ln1_s[tid]; b1[tid] = ln1_b[tid]; }
  __syncthreads();

  int b = blockIdx.x >> 4;
  int chunk = blockIdx.x & 15;
  int t0 = chunk * 128;
  int c = tid & 63, tg = tid >> 6;

  float acc = 0.f;
  float row[32];
  for (int i = 0; i < 32; ++i) {
    conv_relu_ln_row(X, Kc, cb, s1, b1, b, t0 + tg + 4 * i, c, row);
    float rsum = 0.f;
#pragma unroll
    for (int f = 0; f < 32; ++f) rsum += row[f];
    acc += rsum;
  }
  part[tg][c] = acc;
  __syncthreads();
  if (tid < 64)
    chanPart[chunk * 1024 + b * 64 + tid] =
        part[0][tid] + part[1][tid] + part[2][tid] + part[3][tid];
}

// ---- K2: chan = sigmoid(mean @ attn_W + b); G[b][m][n] = Asym[m,n]*chan[b,n]
__global__ void k2_prep(const float* __restrict__ chanPart, const float* __restrict__ attn_W,
                        const float* __restrict__ attn_b, const float* __restrict__ adj,
                        float* __restrict__ G)
{
  __shared__ float meanL[1024], sigL[1024], AsymL[4096];
  int tid = threadIdx.x;
  for (int i = tid; i < 1024; i += 256) {
    float s = 0.f;
    for (int k = 0; k < 16; ++k) s += chanPart[k * 1024 + i];
    meanL[i] = s * (1.0f / 65536.0f);  // / (2048*32)
  }
  for (int i = tid; i < 4096; i += 256) {
    int m = i >> 6, n = i & 63;
    AsymL[i] = (m == n) ? 0.f : 0.5f * (adj[m * 64 + n] + adj[n * 64 + m]);
  }
  __syncthreads();
  for (int i = tid; i < 1024; i += 256) {
    int b = i >> 6, c = i & 63;
    float a = attn_b[c];
    for (int m = 0; m < 64; ++m) a = fmaf(meanL[b * 64 + m], attn_W[m * 64 + c], a);
    sigL[i] = 1.0f / (1.0f + expf(-a));
  }
  __syncthreads();
  for (int i = tid; i < 65536; i += 256) {
    int b = i >> 12, r = i & 4095, m = r >> 6, n = r & 63;
    G[i] = AsymL[m * 64 + n] * sigL[b * 64 + n];
  }
}

// ---- K2b: W_eff[b] = W + (G[b] @ W_f) per f-slice, via fp32 WMMA --------
__global__ void k2b_weff(const float* __restrict__ G, const float* __restrict__ mix_W,
                         float* __restrict__ Weff)
{
  int b = blockIdx.x >> 5;
  int f = blockIdx.x & 31;
  int tid = threadIdx.x;
  int wave = tid >> 5, lane = tid & 31;
  int kh = lane >> 4, l15 = lane & 15;
  const float* Gb = G + b * 4096;
  float* Wo = Weff + (size_t)b * 262144;
  int c0 = wave * 32 + l15;
  int c1 = c0 + 16;

  for (int mt = 0; mt < 4; ++mt) {
    f32x8 accA, accB;
#pragma unroll
    for (int v = 0; v < 8; ++v) {
      int m = mt * 16 + v + 8 * kh;
      accA[v] = mix_W[(m * 32 + f) * 128 + c0];
      accB[v] = mix_W[(m * 32 + f) * 128 + c1];
    }
    int mrow = mt * 16 + l15;
#pragma unroll
    for (int kk = 0; kk < 64; kk += 4) {
      int k0 = kk + 2 * kh;
      f32x2 a, bA, bB;
      a.x  = Gb[mrow * 64 + k0];
      a.y  = Gb[mrow * 64 + k0 + 1];
      bA.x = mix_W[(k0 * 32 + f) * 128 + c0];
      bA.y = mix_W[((k0 + 1) * 32 + f) * 128 + c0];
      bB.x = mix_W[(k0 * 32 + f) * 128 + c1];
      bB.y = mix_W[((k0 + 1) * 32 + f) * 128 + c1];
      accA = WMMA_F32(a, bA, accA);
      accB = WMMA_F32(a, bB, accB);
    }
#pragma unroll
    for (int v = 0; v < 8; ++v) {
      int m = mt * 16 + v + 8 * kh;
      Wo[(m * 32 + f) * 128 + c0] = accA[v];
      Wo[(m * 32 + f) * 128 + c1] = accB[v];
    }
  }
}

// ---- K3: fused conv+LN tile -> GEMM(16,128,2048) w/ W_eff -> relu+LN(128)
// block = (b, t-tile of 16), 128 threads (4 wave32s). B-operand chunks are
// DMA'd into double-buffered LDS by the Tensor Data Mover (TENSORcnt).
__global__ void k3_main(const float* __restrict__ X, const float* __restrict__ conv_k,
                        const float* __restrict__ conv_b, const float* __restrict__ ln1_s,
                        const float* __restrict__ ln1_b, const float* __restrict__ Weff,
                        const float* __restrict__ mix_b, const float* __restrict__ ln2_s,
                        const float* __restrict__ ln2_b, float* __restrict__ Xm)
{
  extern __shared__ float smem[];
  float* XcT  = smem;                    // 16 * 2052
  float* Bc0  = XcT + 16 * XCT_STRIDE;   // 64 * 136
  float* Bc1  = Bc0 + 64 * BC_STRIDE;    // 64 * 136
  float* Kc   = Bc1 + 64 * BC_STRIDE;    // 160
  float* cb   = Kc + 160;                // 32
  float* s1   = cb + 32;                 // 32
  float* b1   = s1 + 32;                 // 32
  float* XmT  = b1 + 32;                 // 16 * 132
  float* stat = XmT + 16 * 132;          // 32

  int tid = threadIdx.x;
  int b   = blockIdx.x >> 7;
  int t0  = (blockIdx.x & 127) * 16;

  for (int i = tid; i < 160; i += 128) Kc[i] = conv_k[i];
  if (tid < 32) { cb[tid] = conv_b[tid]; s1[tid] = ln1_s[tid]; b1[tid] = ln1_b[tid]; }
  __syncthreads();

  // Phase 1: Xc tile (16 t x 64 c x 32 f) -> LDS, layout [t][c*32+f]
  {
    float row[32];
    for (int r = tid; r < 1024; r += 128) {
      int tl = r >> 6, c = r & 63;
      conv_relu_ln_row(X, Kc, cb, s1, b1, b, t0 + tl, c, row);
#pragma unroll
      for (int f = 0; f < 32; ++f) XcT[tl * XCT_STRIDE + c * 32 + f] = row[f];
    }
  }
  __syncthreads();

  // Phase 2: fp32 WMMA GEMM, A = Xc tile (LDS), B = W_eff chunks (LDS, TDM)
  int wave = tid >> 5, lane = tid & 31, kh = lane >> 4, l15 = lane & 15;
  int c0 = wave * 32 + l15, c1 = c0 + 16;
  f32x8 acc0 = {}, acc1 = {};
  const float* Wb = Weff + (size_t)b * 262144;

#if USE_TDM
  unsigned int bcoff0 = (unsigned int)(uintptr_t)Bc0;  // flat->LDS byte offset
  unsigned int bcoff1 = (unsigned int)(uintptr_t)Bc1;
  if (wave == 0) tdm_load_chunk(Wb, bcoff0);           // prefetch chunk 0
#endif

  for (int ci = 0; ci < 32; ++ci) {
    int kb = ci << 6;
    float* Bc = (ci & 1) ? Bc1 : Bc0;
#if USE_TDM
    if (wave == 0) {
      if (ci + 1 < 32) {
        // issue next chunk into the other buffer, then wait for current one
        tdm_load_chunk(Wb + (size_t)(kb + 64) * 128, (ci & 1) ? bcoff0 : bcoff1);
        __builtin_amdgcn_s_wait_tensorcnt(1);   // in-order: current chunk done
      } else {
        __builtin_amdgcn_s_wait_tensorcnt(0);
      }
    }
    __syncthreads();   // data-ready for all waves
#else
    for (int i = tid; i < 2048; i += 128) {             // fallback: float4 copy
      int rowk = i >> 5, col4 = (i & 31) << 2;
      *(float4*)&Bc[rowk * BC_STRIDE + col4] =
          *(const float4*)&Wb[(size_t)(kb + rowk) * 128 + col4];
    }
    __syncthreads();
#endif
#pragma unroll
    for (int kk = 0; kk < 64; kk += 4) {
      int k0 = kk + 2 * kh;
      f32x2 a = *(const f32x2*)&XcT[l15 * XCT_STRIDE + kb + k0];
      f32x2 bv0, bv1;
      bv0.x = Bc[k0 * BC_STRIDE + c0];
      bv0.y = Bc[(k0 + 1) * BC_STRIDE + c0];
      bv1.x = Bc[k0 * BC_STRIDE + c1];
      bv1.y = Bc[(k0 + 1) * BC_STRIDE + c1];
      acc0 = WMMA_F32(a, bv0, acc0);
      acc1 = WMMA_F32(a, bv1, acc1);
    }
    __syncthreads();   // compute done -> buffer may be overwritten next iter
  }

  // Epilogue: +bias, relu -> LDS; LN over 128; store Xm
#pragma unroll
  for (int v = 0; v < 8; ++v) {
    int tl = v + 8 * kh;
    float v0 = acc0[v] + mix_b[c0]; v0 = v0 > 0.f ? v0 : 0.f;
    float v1 = acc1[v] + mix_b[c1]; v1 = v1 > 0.f ? v1 : 0.f;
    XmT[tl * 132 + c0] = v0;
    XmT[tl * 132 + c1] = v1;
  }
  __syncthreads();
  if (tid < 16) {
    float s = 0.f, q = 0.f;
    for (int j = 0; j < 128; ++j) { float v = XmT[tid * 132 + j]; s += v; q = fmaf(v, v, q); }
    float mu  = s * (1.f / 128.f);
    float var = q * (1.f / 128.f) - mu * mu;
    stat[tid * 2] = mu;
    stat[tid * 2 + 1] = rsqrtf(var + 1e-6f);
  }
  __syncthreads();
  {
    int tl = tid >> 3, jb = (tid & 7) * 16;
    float mu = stat[tl * 2], rs = stat[tl * 2 + 1];
    float* dst = Xm + ((size_t)(b * 2048 + t0 + tl)) * 128 + jb;
#pragma unroll
    for (int g = 0; g < 4; ++g) {
      int j = jb + g * 4;
      float4 o;
      o.x = (XmT[tl * 132 + j + 0] - mu) * rs * ln2_s[j + 0] + ln2_b[j + 0];
      o.y = (XmT[tl * 132 + j + 1] - mu) * rs * ln2_s[j + 1] + ln2_b[j + 1];
      o.z = (XmT[tl * 132 + j + 2] - mu) * rs * ln2_s[j + 2] + ln2_b[j + 2];
      o.w = (XmT[tl * 132 + j + 3] - mu) * rs * ln2_s[j + 3] + ln2_b[j + 3];
      *(float4*)&dst[g * 4] = o;
    }
  }
}

// ---- K4: LIF scan (serial over t, parallel over (b,j)) ------------------
__global__ void k4_lif(const float* __restrict__ Xm, float* __restrict__ featSum,
                       float* __restrict__ spkTot)
{
  int b = blockIdx.x, j = threadIdx.x;
  const float* p = Xm + (size_t)b * 2048 * 128 + j;
  float mem = 0.f, cnt = 0.f;
  for (int t = 0; t < 2048; ++t) {
    mem = mem * 0.9f + p[(size_t)t * 128];
    float s = (mem > 0.5f) ? 1.f : 0.f;
    mem -= 0.5f * s;
    cnt += s;
  }
  featSum[b * 128 + j] = cnt * (1.f / 2048.f);
  atomicAdd(spkTot, cnt);  // integer-valued f32 -> order-independent, exact
}

// ---- K5: heads ----------------------------------------------------------
__global__ void k5_head(const float* __restrict__ featSum, const float* __restrict__ fc_W,
                        const float* __restrict__ fc_b, const float* __restrict__ out_W,
                        const float* __restrict__ out_b, const float* __restrict__ spkTot,
                        float* __restrict__ out)
{
  __shared__ float featL[2048], hL[2048];
  int tid = threadIdx.x;
  for (int i = tid; i < 2048; i += 128) featL[i] = featSum[i];
  __syncthreads();
  for (int b = 0; b < 16; ++b) {
    float a = fc_b[tid];
    for (int k = 0; k < 128; ++k) a = fmaf(featL[b * 128 + k], fc_W[k * 128 + tid], a);
    hL[b * 128 + tid] = a > 0.f ? a : 0.f;
  }
  __syncthreads();
  if (tid < 64) {
    int b = tid >> 2, o = tid & 3;
    float a = out_b[o];
    for (int k = 0; k < 128; ++k) a = fmaf(hL[b * 128 + k], out_W[k * 4 + o], a);
    out[b * 4 + o] = a;
  }
  if (tid == 127) out[64] = spkTot[0] * (1.0f / (16.0f * 2048.0f * 128.0f));
}

extern "C" void kernel_launch(void* const* d_in, const int* in_sizes, int n_in,
                              void* d_out, int out_size, void* d_ws, size_t ws_size,
                              hipStream_t stream)
{
  (void)in_sizes; (void)n_in; (void)out_size; (void)ws_size;
  const float* X      = (const float*)d_in[0];
  const float* conv_k = (const float*)d_in[1];
  const float* conv_b = (const float*)d_in[2];
  const float* ln1_s  = (const float*)d_in[3];
  const float* ln1_b  = (const float*)d_in[4];
  const float* attn_W = (const float*)d_in[5];
  const float* attn_b = (const float*)d_in[6];
  const float* adj    = (const float*)d_in[7];
  const float* mix_W  = (const float*)d_in[8];
  const float* mix_b  = (const float*)d_in[9];
  const float* ln2_s  = (const float*)d_in[10];
  const float* ln2_b  = (const float*)d_in[11];
  const float* fc_W   = (const float*)d_in[12];
  const float* fc_b   = (const float*)d_in[13];
  const float* out_W  = (const float*)d_in[14];
  const float* out_b  = (const float*)d_in[15];

  float* ws       = (float*)d_ws;
  float* chanPart = ws;                   // 16 * 1024
  float* G        = chanPart + 16384;     // 16 * 64 * 64
  float* Weff     = G + 65536;            // 16 * 2048 * 128
  float* Xm       = Weff + 4194304;       // 16 * 2048 * 128
  float* featSum  = Xm + 4194304;         // 16 * 128
  float* spkTot   = featSum + 2048;       // 1   (total ~33.9 MB)

  hipMemsetAsync(spkTot, 0, sizeof(float), stream);

  k1_chan<<<256, 256, 0, stream>>>(X, conv_k, conv_b, ln1_s, ln1_b, chanPart);
  k2_prep<<<1, 256, 0, stream>>>(chanPart, attn_W, attn_b, adj, G);
  k2b_weff<<<512, 128, 0, stream>>>(G, mix_W, Weff);
  size_t smem3 = (size_t)(16 * XCT_STRIDE + 2 * 64 * BC_STRIDE + 256 + 16 * 132 + 32) *
                 sizeof(float);
  k3_main<<<2048, 128, smem3, stream>>>(X, conv_k, conv_b, ln1_s, ln1_b, Weff,
                                        mix_b, ln2_s, ln2_b, Xm);
  k4_lif<<<16, 128, 0, stream>>>(Xm, featSum, spkTot);
  k5_head<<<1, 128, 0, stream>>>(featSum, fc_W, fc_b, out_W, out_b, spkTot, (float*)d_out);
}